// HAM_GATConv_27745488732228
// MI455X (gfx1250) — compile-verified
//
#include <hip/hip_runtime.h>
#include <hip/hip_bf16.h>

typedef __attribute__((ext_vector_type(2))) float v2f;
typedef __attribute__((ext_vector_type(8))) float v8f;

#define HEADS      8
#define D_K        16
#define FEAT       128
#define LRELU_A    0.2f

#define TILE_ROWS  16
#define LDS_PITCH  132           // 128 + 4 pad -> conflict-free ds_load_b64 across 16 rows
#define WAVES_PB   4             // 4 waves (wave32) per block, 33 KB LDS

// ---------------- init: segmax = -inf, denom = 0 ----------------
__global__ void gat_init(float* __restrict__ segmax, float* __restrict__ denom, int n) {
  int i = blockIdx.x * blockDim.x + threadIdx.x;
  if (i < n) {
    segmax[i] = -__builtin_inff();
    denom[i]  = 0.0f;
  }
}

// ---------------- node scores via V_WMMA_F32_16X16X4_F32 ----------------
// snode[n*16 + h]     = sum_k a1[k] * x[n, h*16+k]   (h = 0..7)
// snode[n*16 + 8 + h] = sum_k a2[k] * x[n, h*16+k]
// S(16x16) = Xtile(16x128) * W(128x16), W[c, j] sparse from a1/a2; K-loop in steps of 4.
__global__ __launch_bounds__(WAVES_PB * 32)
void gat_node_scores(const float* __restrict__ x, const float* __restrict__ a,
                     float* __restrict__ snode, int nNodes) {
  __shared__ float lds[WAVES_PB][TILE_ROWS * LDS_PITCH];
  const int wave   = threadIdx.x >> 5;
  const int lane   = threadIdx.x & 31;
  const int tile   = blockIdx.x * WAVES_PB + wave;
  const int nTiles = (nNodes + TILE_ROWS - 1) / TILE_ROWS;
  const int m      = lane & 15;   // A-row (M) == B/D column (N) for this lane
  const int khalf  = lane >> 4;   // lane-half selects K pair {0,1}/{2,3} per ISA layout

  if (tile < nTiles) {
    // stage 16x128 f32 tile: fully coalesced b128 loads, one row per iteration
    for (int i = 0; i < TILE_ROWS; ++i) {
      int row = tile * TILE_ROWS + i;
      float4 v = make_float4(0.f, 0.f, 0.f, 0.f);
      if (row < nNodes) v = ((const float4*)(x + (size_t)row * FEAT))[lane];
      ((float4*)(&lds[wave][i * LDS_PITCH]))[lane] = v;
    }
  }
  __syncthreads();
  if (tile >= nTiles) return;

  // W row index within a K=4 slice is k = r + 2*khalf; global column
  // c = head*16 + 4*q + k, so the a-index (c & 15) = 4*q + 2*khalf + r.
  float a1v[2][4], a2v[2][4];
#pragma unroll
  for (int r = 0; r < 2; ++r)
#pragma unroll
    for (int q = 0; q < 4; ++q) {
      int idx = 4 * q + 2 * khalf + r;
      a1v[r][q] = a[idx];
      a2v[r][q] = a[D_K + idx];
    }

  v8f acc = {};
  const float* tl = &lds[wave][0];
#pragma unroll
  for (int head = 0; head < HEADS; ++head) {
#pragma unroll
    for (int q = 0; q < 4; ++q) {
      const int kcol = head * 16 + q * 4 + 2 * khalf;
      v2f A;  // A[m, k]: VGPR r holds K = r + 2*khalf  -> ds_load_b64 (pitch-padded)
      A.x = tl[m * LDS_PITCH + kcol];
      A.y = tl[m * LDS_PITCH + kcol + 1];
      v2f B;  // B[k, n]: column n==head gets a1, n==head+8 gets a2, else 0
      B.x = (m == head) ? a1v[0][q] : ((m == head + 8) ? a2v[0][q] : 0.0f);
      B.y = (m == head) ? a1v[1][q] : ((m == head + 8) ? a2v[1][q] : 0.0f);
      acc = __builtin_amdgcn_wmma_f32_16x16x4_f32(false, A, false, B,
                                                  (short)0, acc, false, false);
    }
  }

  // D layout: VGPR r -> node row r (lanes 0-15) / r+8 (lanes 16-31), column = m
#pragma unroll
  for (int r = 0; r < 8; ++r) {
    int node = tile * TILE_ROWS + r + 8 * khalf;
    if (node < nNodes) snode[(size_t)node * 16 + m] = acc[r];
  }
}

// ---------------- float atomic max (mixed-sign-safe bit trick) ----------------
__device__ __forceinline__ void atomicMaxF32(float* addr, float val) {
  if (val >= 0.0f)
    atomicMax((int*)addr, __float_as_int(val));
  else
    atomicMin((unsigned int*)addr, __float_as_uint(val));
}

__device__ __forceinline__ void gatherScores(const float* __restrict__ snode,
                                             long long s, long long d, float* sv) {
  const float4* p1 = (const float4*)(snode + (size_t)s * 16);      // a1 . h[src]
  const float4* p2 = (const float4*)(snode + (size_t)d * 16 + 8);  // a2 . h[dst]
  float4 u0 = p1[0], u1 = p1[1], w0 = p2[0], w1 = p2[1];
  sv[0] = u0.x + w0.x; sv[1] = u0.y + w0.y; sv[2] = u0.z + w0.z; sv[3] = u0.w + w0.w;
  sv[4] = u1.x + w1.x; sv[5] = u1.y + w1.y; sv[6] = u1.z + w1.z; sv[7] = u1.w + w1.w;
}

// ---------------- pass 1 over edges: segment max ----------------
__global__ void gat_edge_max(const float* __restrict__ snode,
                             const long long* __restrict__ edge,
                             float* __restrict__ segmax, int nE) {
  int e = blockIdx.x * blockDim.x + threadIdx.x;
  if (e >= nE) return;
  long long s = edge[e], d = edge[(size_t)nE + e];
  float sv[8];
  gatherScores(snode, s, d, sv);
  float* base = segmax + (size_t)d * HEADS;
#pragma unroll
  for (int h = 0; h < HEADS; ++h) {
    float ev = sv[h] > 0.0f ? sv[h] : LRELU_A * sv[h];
    atomicMaxF32(base + h, ev);
  }
}

// ---------------- pass 2: ex = exp(e - max), denom += ex, out = ex ----------------
__global__ void gat_edge_exp(const float* __restrict__ snode,
                             const long long* __restrict__ edge,
                             const float* __restrict__ segmax,
                             float* __restrict__ denom,
                             float* __restrict__ out, int nE) {
  int e = blockIdx.x * blockDim.x + threadIdx.x;
  if (e >= nE) return;
  long long s = edge[e], d = edge[(size_t)nE + e];
  float sv[8];
  gatherScores(snode, s, d, sv);
  const float4* mx = (const float4*)(segmax + (size_t)d * HEADS);
  float4 m0 = mx[0], m1 = mx[1];
  float mv[8] = {m0.x, m0.y, m0.z, m0.w, m1.x, m1.y, m1.z, m1.w};
  float ex[8];
  float* dn = denom + (size_t)d * HEADS;
#pragma unroll
  for (int h = 0; h < HEADS; ++h) {
    float ev = sv[h] > 0.0f ? sv[h] : LRELU_A * sv[h];
    ex[h] = expf(ev - mv[h]);
    atomicAdd(dn + h, ex[h]);
  }
  float4* o = (float4*)(out + (size_t)e * HEADS);
  o[0] = make_float4(ex[0], ex[1], ex[2], ex[3]);
  o[1] = make_float4(ex[4], ex[5], ex[6], ex[7]);
}

// ---------------- pass 3: normalize ----------------
__global__ void gat_norm(const long long* __restrict__ edge,
                         const float* __restrict__ denom,
                         float* __restrict__ out, int nE) {
  int e = blockIdx.x * blockDim.x + threadIdx.x;
  if (e >= nE) return;
  long long d = edge[(size_t)nE + e];
  const float4* dn = (const float4*)(denom + (size_t)d * HEADS);
  float4 d0 = dn[0], d1 = dn[1];
  float4* o = (float4*)(out + (size_t)e * HEADS);
  float4 o0 = o[0], o1 = o[1];
  o0.x /= (d0.x + 1e-16f); o0.y /= (d0.y + 1e-16f);
  o0.z /= (d0.z + 1e-16f); o0.w /= (d0.w + 1e-16f);
  o1.x /= (d1.x + 1e-16f); o1.y /= (d1.y + 1e-16f);
  o1.z /= (d1.z + 1e-16f); o1.w /= (d1.w + 1e-16f);
  o[0] = o0; o[1] = o1;
}

extern "C" void kernel_launch(void* const* d_in, const int* in_sizes, int n_in,
                              void* d_out, int out_size, void* d_ws, size_t ws_size,
                              hipStream_t stream) {
  const float*     x    = (const float*)d_in[0];      // (N, 128) f32
  const float*     a    = (const float*)d_in[1];      // (32,) f32
  const long long* edge = (const long long*)d_in[2];  // (2, E) i64
  float*           out  = (float*)d_out;              // (E, 8) f32

  const int nNodes = in_sizes[0] / FEAT;
  const int nE     = in_sizes[2] / 2;

  // workspace: snode (N*16 f32) | segmax (N*8 f32) | denom (N*8 f32) = 128*N bytes... 12.8 MB
  float* snode  = (float*)d_ws;
  float* segmax = snode  + (size_t)nNodes * 16;
  float* denom  = segmax + (size_t)nNodes * HEADS;

  const int initN = nNodes * HEADS;
  gat_init<<<(initN + 255) / 256, 256, 0, stream>>>(segmax, denom, initN);

  const int nTiles = (nNodes + TILE_ROWS - 1) / TILE_ROWS;
  gat_node_scores<<<(nTiles + WAVES_PB - 1) / WAVES_PB, WAVES_PB * 32, 0, stream>>>(
      x, a, snode, nNodes);

  const int eBlocks = (nE + 255) / 256;
  gat_edge_max<<<eBlocks, 256, 0, stream>>>(snode, edge, segmax, nE);
  gat_edge_exp<<<eBlocks, 256, 0, stream>>>(snode, edge, segmax, denom, out, nE);
  gat_norm<<<eBlocks, 256, 0, stream>>>(edge, denom, out, nE);
}